// RCNN_59717225284162
// MI455X (gfx1250) — compile-verified
//
#include <hip/hip_runtime.h>
#include <math.h>

#define DEV __device__ __forceinline__

typedef __attribute__((ext_vector_type(2))) float v2f;
typedef __attribute__((ext_vector_type(8))) float v8f;

constexpr int B = 4;
constexpr int N = 16384;
constexpr int C = 76;
constexpr int NB = 12;           // loc bins per axis
constexpr int NHB = 12;          // heading bins
constexpr int PRE_NEAR = 6300;
constexpr int PRE_FAR  = 2700;
constexpr int POST_NEAR = 358;
constexpr int POST_FAR  = 154;
constexpr int NMS_POST  = 512;
constexpr int TCH = 32;          // greedy NMS chunk == one wave32
constexpr float NMS_THRES = 0.8f;
constexpr float LOC_SCOPE = 3.0f;
constexpr float LOC_BIN   = 0.5f;
constexpr float PI_F      = 3.14159265358979323846f;
constexpr float TWO_PI_F  = 6.28318530717958647692f;

// ---------------- helpers ----------------

DEV float area4(const float* bv) { return (bv[2] - bv[0]) * (bv[3] - bv[1]); }

// exclusive block scan of one unsigned per thread; tmp has blockDim.x entries
DEV unsigned block_scan_excl(unsigned v, unsigned* tmp) {
  const int tid = threadIdx.x;
  const int n = blockDim.x;
  tmp[tid] = v;
  __syncthreads();
  for (int off = 1; off < n; off <<= 1) {
    unsigned x = (tid >= off) ? tmp[tid - off] : 0u;
    __syncthreads();
    tmp[tid] += x;
    __syncthreads();
  }
  unsigned incl = tmp[tid];
  __syncthreads();
  return incl - v;
}

// ---------------- kernel 1: decode + sort keys ----------------

__global__ void k_decode(const float* __restrict__ scores,
                         const float* __restrict__ reg,
                         const float* __restrict__ xyz,
                         const float* __restrict__ anchor,
                         float* __restrict__ data0,
                         unsigned long long* __restrict__ keys) {
  int gid = blockIdx.x * blockDim.x + threadIdx.x;
  if (gid >= B * N) return;
  const float* r = reg + (size_t)gid * C;
  const float* roi = xyz + (size_t)gid * 7;

  int xb = 0; float xv = r[0];
  for (int t = 1; t < NB; ++t) { float v = r[t]; if (v > xv) { xv = v; xb = t; } }
  int zb = 0; float zv = r[NB];
  for (int t = 1; t < NB; ++t) { float v = r[NB + t]; if (v > zv) { zv = v; zb = t; } }
  float x_res = r[2 * NB + xb] * LOC_BIN;
  float z_res = r[3 * NB + zb] * LOC_BIN;
  float pos_x = (float)xb * LOC_BIN + LOC_BIN * 0.5f - LOC_SCOPE + x_res;
  float pos_z = (float)zb * LOC_BIN + LOC_BIN * 0.5f - LOC_SCOPE + z_res;

  int off = 4 * NB;
  float pos_y = roi[1] + r[off];
  off += 1;
  int rb = 0; float rv = r[off];
  for (int t = 1; t < NHB; ++t) { float v = r[off + t]; if (v > rv) { rv = v; rb = t; } }
  float ry_res = r[off + NHB + rb];
  const float apc = TWO_PI_F / (float)NHB;
  float ry = (float)rb * apc + ry_res * (apc * 0.5f);
  ry = fmodf(ry, TWO_PI_F);
  if (ry < 0.f) ry += TWO_PI_F;      // floor-mod -> [0, 2pi)
  if (ry > PI_F) ry -= TWO_PI_F;
  off += 2 * NHB;

  float a0 = anchor[0], a1 = anchor[1], a2 = anchor[2];
  float h = r[off + 0] * a0 + a0;
  float w = r[off + 1] * a1 + a1;
  float l = r[off + 2] * a2 + a2;

  float roi_ry = roi[6];
  float cs = cosf(roi_ry), sn = sinf(roi_ry);
  float xr =  pos_x * cs + pos_z * sn + roi[0];
  float zr = -pos_x * sn + pos_z * cs + roi[2];
  ry += roi_ry;
  float y_out = pos_y + 0.5f * h;    // props[...,1] += props[...,3]/2

  float s = scores[gid];
  float* d = data0 + (size_t)gid * 8;
  d[0] = s; d[1] = xr; d[2] = y_out; d[3] = zr;
  d[4] = h; d[5] = w;  d[6] = l;     d[7] = ry;

  unsigned u = __float_as_uint(s);
  unsigned asc = (u & 0x80000000u) ? ~u : (u | 0x80000000u);
  unsigned dk = ~asc;                // ascending dk == descending score
  keys[gid] = ((unsigned long long)dk << 32) | (unsigned)(gid % N);
}

// ---------------- kernel 2: per-batch bitonic sort in LDS (128 KB) ----------------

__global__ void k_sort(const unsigned long long* __restrict__ keys,
                       unsigned* __restrict__ order) {
  extern __shared__ unsigned long long sk[];   // N entries = 128 KB (CDNA5: 320 KB LDS/WGP)
  const int b = blockIdx.x;
  const int tid = threadIdx.x;
  for (int p = tid; p < N; p += blockDim.x) sk[p] = keys[(size_t)b * N + p];
  __syncthreads();
  for (unsigned k = 2; k <= (unsigned)N; k <<= 1) {
    for (unsigned j = k >> 1; j > 0; j >>= 1) {
      for (unsigned i = tid; i < (unsigned)N; i += blockDim.x) {
        unsigned l = i ^ j;
        if (l > i) {
          unsigned long long a = sk[i], c = sk[l];
          bool asc = ((i & k) == 0);
          if ((a > c) == asc) { sk[i] = c; sk[l] = a; }
        }
      }
      __syncthreads();
    }
  }
  for (int p = tid; p < N; p += blockDim.x)
    order[(size_t)b * N + p] = (unsigned)(sk[p] & 0xFFFFFFFFu);
}

// ---------------- kernel 3: distance-split compaction (preK) ----------------

__global__ void k_compact(const float* __restrict__ data0,
                          const unsigned* __restrict__ order,
                          float* __restrict__ cnear, float* __restrict__ cfar,
                          unsigned* __restrict__ vnear, unsigned* __restrict__ vfar) {
  extern __shared__ unsigned ctmp[];           // blockDim.x
  const int b = blockIdx.x, tid = threadIdx.x;
  float* cn = cnear + (size_t)b * PRE_NEAR * 8;
  float* cf = cfar  + (size_t)b * PRE_FAR * 8;
  unsigned* vn = vnear + (size_t)b * PRE_NEAR;
  unsigned* vf = vfar  + (size_t)b * PRE_FAR;
  for (int p = tid; p < PRE_NEAR * 8; p += blockDim.x) cn[p] = 0.f;
  for (int p = tid; p < PRE_FAR * 8;  p += blockDim.x) cf[p] = 0.f;
  for (int p = tid; p < PRE_NEAR; p += blockDim.x) vn[p] = 0u;
  for (int p = tid; p < PRE_FAR;  p += blockDim.x) vf[p] = 0u;

  const int CH = N / 1024;                     // 16 contiguous sorted rows / thread
  const int base = tid * CH;
  unsigned nearBits = 0, farBits = 0;
  for (int kk = 0; kk < CH; ++kk) {
    unsigned oi = order[(size_t)b * N + base + kk];
    float dist = data0[((size_t)b * N + oi) * 8 + 3];   // decoded z (col 3 of [s|prop])
    if (dist > 0.f  && dist <= 40.f) nearBits |= (1u << kk);
    if (dist > 40.f && dist <= 80.f) farBits  |= (1u << kk);
  }
  unsigned nearBase = block_scan_excl((unsigned)__popc(nearBits), ctmp);
  unsigned farBase  = block_scan_excl((unsigned)__popc(farBits),  ctmp);
  for (int kk = 0; kk < CH; ++kk) {
    unsigned oi = order[(size_t)b * N + base + kk];
    const float* row = data0 + ((size_t)b * N + oi) * 8;
    unsigned mlow = (1u << kk) - 1u;
    if (nearBits & (1u << kk)) {
      unsigned rank = nearBase + (unsigned)__popc(nearBits & mlow);
      if (rank < PRE_NEAR) { for (int q = 0; q < 8; ++q) cn[rank * 8 + q] = row[q]; vn[rank] = 1u; }
    }
    if (farBits & (1u << kk)) {
      unsigned rank = farBase + (unsigned)__popc(farBits & mlow);
      if (rank < PRE_FAR) { for (int q = 0; q < 8; ++q) cf[rank * 8 + q] = row[q]; vf[rank] = 1u; }
    }
  }
}

// ---------------- kernel 4: chunked greedy NMS (wave-synchronous chunk + WMMA cross) --

__global__ void k_nms(const float* __restrict__ cnear, const float* __restrict__ cfar,
                      const unsigned* __restrict__ vnear, const unsigned* __restrict__ vfar,
                      float* __restrict__ knear, float* __restrict__ kfar,
                      unsigned* __restrict__ kvnear, unsigned* __restrict__ kvfar) {
  extern __shared__ float smemf[];
  const int rng = blockIdx.x;                  // 0 = near, 1 = far
  const int b = blockIdx.y;
  const int K     = (rng == 0) ? PRE_NEAR : PRE_FAR;
  const int postK = (rng == 0) ? POST_NEAR : POST_FAR;
  const float* cdata   = (rng == 0) ? cnear + (size_t)b * PRE_NEAR * 8 : cfar + (size_t)b * PRE_FAR * 8;
  const unsigned* cval = (rng == 0) ? vnear + (size_t)b * PRE_NEAR     : vfar + (size_t)b * PRE_FAR;
  float* kdata  = (rng == 0) ? knear  + (size_t)b * POST_NEAR * 8 : kfar  + (size_t)b * POST_FAR * 8;
  unsigned* kval = (rng == 0) ? kvnear + (size_t)b * POST_NEAR    : kvfar + (size_t)b * POST_FAR;

  float*    bev   = smemf;                                // PRE_NEAR*4 floats
  unsigned* keepL = (unsigned*)(bev + PRE_NEAR * 4);      // PRE_NEAR
  float*    ibev  = (float*)(keepL + PRE_NEAR);           // TCH*4
  float*    iarea = ibev + TCH * 4;                       // TCH
  unsigned* iList = (unsigned*)(iarea + TCH);             // TCH
  unsigned* stmp  = iList + TCH;                          // 1024
  __shared__ int nI_s;

  const int tid = threadIdx.x;
  for (int p = tid; p < K; p += blockDim.x) {
    const float* row = cdata + (size_t)p * 8;
    float cx = row[1], cz = row[3], w_ = row[4], h_ = row[6];
    bev[p * 4 + 0] = cx - 0.5f * w_;
    bev[p * 4 + 1] = cz - 0.5f * h_;
    bev[p * 4 + 2] = cx + 0.5f * w_;
    bev[p * 4 + 3] = cz + 0.5f * h_;
    keepL[p] = cval[p];
  }
  __syncthreads();

  for (int cStart = 0; cStart < K; cStart += TCH) {
    const int cEnd = (cStart + TCH < K) ? (cStart + TCH) : K;
    // (a) sequential greedy within chunk: wave 0 only, wave-synchronous via shuffles
    if (tid < 32) {
      const int j = cStart + tid;
      const bool inR = (j < cEnd);
      float b0 = 0.f, b1 = 0.f, b2 = 0.f, b3 = 0.f;
      unsigned kf = 0u;
      if (inR) {
        b0 = bev[j * 4 + 0]; b1 = bev[j * 4 + 1];
        b2 = bev[j * 4 + 2]; b3 = bev[j * 4 + 3];
        kf = keepL[j];
      }
      const float aJ = (b2 - b0) * (b3 - b1);
      const int len = cEnd - cStart;
      for (int t = 0; t < len; ++t) {
        unsigned ki = __shfl(kf, t, 32);       // lane t's CURRENT keep flag
        float i0 = __shfl(b0, t, 32), i1 = __shfl(b1, t, 32);
        float i2 = __shfl(b2, t, 32), i3 = __shfl(b3, t, 32);
        if (ki && inR && tid > t) {
          float iw = fminf(i2, b2) - fmaxf(i0, b0); iw = fmaxf(iw, 0.f);
          float ih = fminf(i3, b3) - fmaxf(i1, b1); ih = fmaxf(ih, 0.f);
          float inter = iw * ih;
          float aI = (i2 - i0) * (i3 - i1);
          float uni = aI + aJ - inter;
          if (inter > NMS_THRES * fmaxf(uni, 1e-8f)) kf = 0u;
        }
      }
      if (inR) keepL[j] = kf;
      // (b) collect kept rows of this chunk (same wave: LDS ops are in-order)
      if (tid == 0) {
        int n = 0;
        for (int i = cStart; i < cEnd; ++i) {
          if (keepL[i]) {
            iList[n] = (unsigned)i;
            float x0 = bev[i * 4 + 0], y0 = bev[i * 4 + 1];
            float x1 = bev[i * 4 + 2], y1 = bev[i * 4 + 3];
            ibev[n * 4 + 0] = x0; ibev[n * 4 + 1] = y0;
            ibev[n * 4 + 2] = x1; ibev[n * 4 + 3] = y1;
            iarea[n] = (x1 - x0) * (y1 - y0);
            ++n;
          }
        }
        nI_s = n;
      }
    }
    __syncthreads();
    const int nI = nI_s;                       // block-uniform
    if (nI > 0) {
      const int lane = tid & 31;
      const int wave = tid >> 5;               // wave32 on gfx1250
      const int n16 = lane & 15;
      const bool lowHalf = (lane < 16);
      const int nTilesI = (nI + 15) >> 4;      // <= 2
      const int jstride = (int)(blockDim.x >> 5) * 16;
      for (int jt = cEnd + wave * 16; jt < K; jt += jstride) {
        const int j = jt + n16;
        float jx0 = 0.f, jy0 = 0.f, jx1 = 0.f, jy1 = 0.f, areaJ = 0.f;
        if (j < K) {
          jx0 = bev[j * 4 + 0]; jy0 = bev[j * 4 + 1];
          jx1 = bev[j * 4 + 2]; jy1 = bev[j * 4 + 3];
          areaJ = (jx1 - jx0) * (jy1 - jy0);
        }
        // B (4x16): VGPR0 = {K0 low half, K2 high half}, VGPR1 = {K1, K3}
        v2f bfrag;
        bfrag.x = lowHalf ? 1.0f : 0.0f;       // K0 row = ones
        bfrag.y = lowHalf ? areaJ : 0.0f;      // K1 row = area_j
        for (int it = 0; it < nTilesI; ++it) {
          int ar = it * 16 + n16;
          bool av = lowHalf && (ar < nI);
          // A (16x4): lanes 0-15 hold K0,K1 of row m; lanes 16-31 hold K2,K3 (zero)
          v2f afrag;
          afrag.x = av ? iarea[ar] : 0.0f;     // K0 = area_i
          afrag.y = av ? 1.0f : 0.0f;          // K1 = 1
          v8f cz8 = {};
          // D[m][n] = area_i(m) + area_j(n) on the matrix pipe
          v8f d = __builtin_amdgcn_wmma_f32_16x16x4_f32(
              false, afrag, false, bfrag, (short)0, cz8, false, false);
          const int mBase = lowHalf ? 0 : 8;   // D: VGPR r -> M=r (low) / M=r+8 (high)
          #pragma unroll
          for (int r = 0; r < 8; ++r) {
            int m = it * 16 + mBase + r;
            if (m < nI && j < K) {
              float ix0 = ibev[m * 4 + 0], iy0 = ibev[m * 4 + 1];
              float ix1 = ibev[m * 4 + 2], iy1 = ibev[m * 4 + 3];
              float iw = fminf(ix1, jx1) - fmaxf(ix0, jx0); iw = fmaxf(iw, 0.f);
              float ih = fminf(iy1, jy1) - fmaxf(iy0, jy0); ih = fmaxf(ih, 0.f);
              float inter = iw * ih;
              float uni = d[r] - inter;
              if (inter > NMS_THRES * fmaxf(uni, 1e-8f)) keepL[j] = 0u;
            }
          }
        }
      }
    }
    __syncthreads();
  }

  // compact kept rows -> postK
  for (int p = tid; p < postK * 8; p += blockDim.x) kdata[p] = 0.f;
  for (int p = tid; p < postK; p += blockDim.x) kval[p] = 0u;
  const int CH = (K + (int)blockDim.x - 1) / (int)blockDim.x;   // <= 7
  const int base = tid * CH;
  unsigned bits = 0;
  for (int kk = 0; kk < CH; ++kk) {
    int p = base + kk;
    if (p < K && keepL[p]) bits |= (1u << kk);
  }
  unsigned rbase = block_scan_excl((unsigned)__popc(bits), stmp);
  for (int kk = 0; kk < CH; ++kk) {
    int p = base + kk;
    if (bits & (1u << kk)) {
      unsigned rank = rbase + (unsigned)__popc(bits & ((1u << kk) - 1u));
      if (rank < (unsigned)postK) {
        const float* row = cdata + (size_t)p * 8;
        for (int q = 0; q < 8; ++q) kdata[rank * 8 + q] = row[q];
        kval[rank] = 1u;
      }
    }
  }
}

// ---------------- kernel 5: merge near+far, final compact to 512 ----------------

__global__ void k_finalize(const float* __restrict__ knear, const float* __restrict__ kfar,
                           const unsigned* __restrict__ kvnear, const unsigned* __restrict__ kvfar,
                           float* __restrict__ out) {
  extern __shared__ unsigned ftmp[];           // blockDim.x == 512
  const int b = blockIdx.x, tid = threadIdx.x;
  float* obox   = out + (size_t)b * NMS_POST * 7;
  float* oscore = out + (size_t)B * NMS_POST * 7 + (size_t)b * NMS_POST;
  for (int p = tid; p < NMS_POST * 7; p += blockDim.x) obox[p] = 0.f;
  for (int p = tid; p < NMS_POST; p += blockDim.x) oscore[p] = 0.f;

  unsigned v;
  const float* row;
  if (tid < POST_NEAR) {
    v = kvnear[(size_t)b * POST_NEAR + tid];
    row = knear + ((size_t)b * POST_NEAR + tid) * 8;
  } else {
    int t2 = tid - POST_NEAR;
    v = kvfar[(size_t)b * POST_FAR + t2];
    row = kfar + ((size_t)b * POST_FAR + t2) * 8;
  }
  unsigned rank = block_scan_excl(v, ftmp);
  if (v) {                                     // total valid <= 512 -> rank always in range
    oscore[rank] = row[0];
    for (int q = 0; q < 7; ++q) obox[rank * 7 + q] = row[1 + q];
  }
}

// ---------------- host launch ----------------

extern "C" void kernel_launch(void* const* d_in, const int* in_sizes, int n_in,
                              void* d_out, int out_size, void* d_ws, size_t ws_size,
                              hipStream_t stream) {
  (void)in_sizes; (void)n_in; (void)out_size; (void)ws_size;
  const float* scores = (const float*)d_in[0];
  const float* reg    = (const float*)d_in[1];
  const float* xyz    = (const float*)d_in[2];
  const float* anchor = (const float*)d_in[3];
  float* out = (float*)d_out;

  char* ws = (char*)d_ws;
  size_t off = 0;
  auto alloc = [&](size_t bytes, size_t align) -> char* {
    off = (off + align - 1) & ~(align - 1);
    char* p = ws + off; off += bytes; return p;
  };
  unsigned long long* keys = (unsigned long long*)alloc((size_t)B * N * 8, 8);
  float* data0   = (float*)alloc((size_t)B * N * 8 * 4, 4);
  unsigned* order = (unsigned*)alloc((size_t)B * N * 4, 4);
  float* cnear   = (float*)alloc((size_t)B * PRE_NEAR * 8 * 4, 4);
  float* cfar    = (float*)alloc((size_t)B * PRE_FAR * 8 * 4, 4);
  unsigned* vnear = (unsigned*)alloc((size_t)B * PRE_NEAR * 4, 4);
  unsigned* vfar  = (unsigned*)alloc((size_t)B * PRE_FAR * 4, 4);
  float* knear   = (float*)alloc((size_t)B * POST_NEAR * 8 * 4, 4);
  float* kfar    = (float*)alloc((size_t)B * POST_FAR * 8 * 4, 4);
  unsigned* kvnear = (unsigned*)alloc((size_t)B * POST_NEAR * 4, 4);
  unsigned* kvfar  = (unsigned*)alloc((size_t)B * POST_FAR * 4, 4);

  k_decode<<<(B * N + 255) / 256, 256, 0, stream>>>(scores, reg, xyz, anchor, data0, keys);
  k_sort<<<B, 1024, (size_t)N * sizeof(unsigned long long), stream>>>(keys, order);
  k_compact<<<B, 1024, 1024 * sizeof(unsigned), stream>>>(data0, order, cnear, cfar, vnear, vfar);
  size_t nmsSmem = (size_t)(PRE_NEAR * 4 + TCH * 4 + TCH) * sizeof(float)
                 + (size_t)(PRE_NEAR + TCH + 1024) * sizeof(unsigned);
  k_nms<<<dim3(2, B), 1024, nmsSmem, stream>>>(cnear, cfar, vnear, vfar,
                                               knear, kfar, kvnear, kvfar);
  k_finalize<<<B, 512, 512 * sizeof(unsigned), stream>>>(knear, kfar, kvnear, kvfar, out);
}